// GraphSAGEEncoder_70007966925389
// MI455X (gfx1250) — compile-verified
//
#include <hip/hip_runtime.h>
#include <hip/hip_bf16.h>

// GraphSAGE encoder for MI455X (gfx1250, wave32).
// GEMMs use V_WMMA_F32_16X16X4_F32 (fp32-exact). Aggregation is pushed
// after the linear transform (mean is linear), halving layer-0 edge traffic.
// A-tiles are staged into LDS with GLOBAL_LOAD_ASYNC_TO_LDS_B128 (ASYNCcnt).

typedef float v2f __attribute__((ext_vector_type(2)));
typedef float v8f __attribute__((ext_vector_type(8)));

#define N_NODES 50000
#define G_GRAPHS 64
#define OUT_DIM 128

// ---------------------------------------------------------------------------
// Dual GEMM: P = X @ Wl, R = X @ Wr   (X: [n x K], W: [K x FOUT] row-major)
// One 16-row tile per block, one 16-col tile per wave (FOUT/16 waves).
// A tile staged in LDS via async-to-LDS; +4 float row padding keeps every
// 16B async chunk aligned AND rotates banks by 4/row -> conflict-free
// ds_load_b64 A-fragment reads (lanes m hit banks 4m,4m+1).
// WMMA fragment layout (ISA 7.12.2, 32-bit A 16x4 / B 4x16 / C 16x16):
//   A: lane m=lane&15, k = 2*(lane>>4) + v      (v = vgpr 0/1)
//   B: lane n=lane&15, k = 2*(lane>>4) + v
//   D: vgpr r -> row = r + 8*(lane>>4), col = lane&15
// ---------------------------------------------------------------------------
template <int K, int FOUT>
__global__ __launch_bounds__(FOUT * 2) void sage_dual_gemm(
    const float* __restrict__ X, const float* __restrict__ Wl,
    const float* __restrict__ Wr, float* __restrict__ P, float* __restrict__ R) {
  constexpr int LDA = K + 4;  // pad 4 floats: 16B-aligned rows, 4-bank rotate
  __shared__ float As[16 * LDA];
  const int tid  = threadIdx.x;
  const int row0 = blockIdx.x * 16;

  // Async global->LDS staging of the 16 x K A-tile (16B chunks per lane).
  constexpr int NCHUNK = (16 * K) / 4;
  for (int i = tid; i < NCHUNK; i += FOUT * 2) {
    const int r = (i * 4) / K;
    const int c = (i * 4) % K;
    // Low 32 bits of a generic pointer into LDS == byte offset from LDS base
    // (addrspacecast AS3->generic is {aperture_hi, lds_offset} on AMDGPU).
    const unsigned lds_off =
        (unsigned)(unsigned long long)(const void*)(As + r * LDA + c);
    const float* g = X + (size_t)(row0 + r) * K + c;
    asm volatile("global_load_async_to_lds_b128 %0, %1, off"
                 :
                 : "v"(lds_off), "v"(g)
                 : "memory");
  }
  asm volatile("s_wait_asynccnt 0x0" ::: "memory");
  __syncthreads();

  const int lane = tid & 31;
  const int c0   = (tid >> 5) << 4;     // wave -> column tile
  const int m    = lane & 15;
  const int koff = (lane >> 4) << 1;

  v8f accP = {0.f, 0.f, 0.f, 0.f, 0.f, 0.f, 0.f, 0.f};
  v8f accR = {0.f, 0.f, 0.f, 0.f, 0.f, 0.f, 0.f, 0.f};

#pragma unroll 4
  for (int kb = 0; kb < K; kb += 4) {
    const float* arow = As + m * LDA + kb + koff;   // ds_load_b64
    v2f a;  a[0] = arow[0];  a[1] = arow[1];
    const float* wl = Wl + (size_t)(kb + koff) * FOUT + c0 + m;
    v2f bl; bl[0] = wl[0]; bl[1] = wl[FOUT];        // W is tiny -> L1/L2 hot
    const float* wr = Wr + (size_t)(kb + koff) * FOUT + c0 + m;
    v2f br; br[0] = wr[0]; br[1] = wr[FOUT];
    accP = __builtin_amdgcn_wmma_f32_16x16x4_f32(false, a, false, bl,
                                                 (short)0, accP, false, false);
    accR = __builtin_amdgcn_wmma_f32_16x16x4_f32(false, a, false, br,
                                                 (short)0, accR, false, false);
  }

  const int rbase = row0 + ((lane >> 4) << 3);
#pragma unroll
  for (int r = 0; r < 8; ++r) {
    P[(size_t)(rbase + r) * FOUT + c0 + m] = accP[r];
    R[(size_t)(rbase + r) * FOUT + c0 + m] = accR[r];
  }
}

// ---------------------------------------------------------------------------
__global__ void zero_f32(float* __restrict__ p, long long n) {
  long long i = (long long)blockIdx.x * blockDim.x + threadIdx.x;
  long long stride = (long long)gridDim.x * blockDim.x;
  for (; i < n; i += stride) p[i] = 0.0f;
}

__global__ void deg_accum(const int* __restrict__ edst, float* __restrict__ deg,
                          int num_edges) {
  int e = blockIdx.x * blockDim.x + threadIdx.x;
  if (e < num_edges) atomicAdd(&deg[edst[e]], 1.0f);
}

__global__ void deg_invert(float* __restrict__ deg, int n) {
  int i = blockIdx.x * blockDim.x + threadIdx.x;
  if (i < n) deg[i] = 1.0f / fmaxf(deg[i], 1.0f);
}

// Scatter-add of transformed features over edges: Agg[dst] += P[src].
// One thread per (edge, float4-chunk); qshift = log2(F/4).
__global__ void scatter_accum(const int* __restrict__ esrc,
                              const int* __restrict__ edst,
                              const float* __restrict__ P,
                              float* __restrict__ Agg, int num_edges,
                              int qshift) {
  long long idx = (long long)blockIdx.x * blockDim.x + threadIdx.x;
  long long total = ((long long)num_edges) << qshift;
  if (idx >= total) return;
  const int e = (int)(idx >> qshift);
  const int q = (int)(idx & ((1 << qshift) - 1));
  const int F = 4 << qshift;
  const int s = esrc[e], d = edst[e];
  const float4 v =
      *reinterpret_cast<const float4*>(P + (size_t)s * F + (size_t)q * 4);
  float* a = Agg + (size_t)d * F + (size_t)q * 4;
  atomicAdd(a + 0, v.x);
  atomicAdd(a + 1, v.y);
  atomicAdd(a + 2, v.z);
  atomicAdd(a + 3, v.w);
}

// h = Agg*invdeg + R + bias ; optional BN(eval) + ReLU.
// float4-vectorized: one thread per (node, float4 chunk); qshift=log2(F/4).
__global__ void ew_bn_relu4(const float* __restrict__ Agg,
                            const float* __restrict__ Rm,
                            const float* __restrict__ invdeg,
                            const float* __restrict__ bias,
                            const float* __restrict__ gamma,
                            const float* __restrict__ beta,
                            const float* __restrict__ mean,
                            const float* __restrict__ var,
                            float* __restrict__ out, int n_nodes, int qshift,
                            int do_bn_relu) {
  long long idx = (long long)blockIdx.x * blockDim.x + threadIdx.x;
  long long total = ((long long)n_nodes) << qshift;
  if (idx >= total) return;
  const int node = (int)(idx >> qshift);
  const int q = (int)(idx & ((1 << qshift) - 1));
  const float id = invdeg[node];

  const float4 ag = reinterpret_cast<const float4*>(Agg)[idx];
  const float4 rr = reinterpret_cast<const float4*>(Rm)[idx];
  const float4 bb = reinterpret_cast<const float4*>(bias)[q];
  float4 h;
  h.x = ag.x * id + rr.x + bb.x;
  h.y = ag.y * id + rr.y + bb.y;
  h.z = ag.z * id + rr.z + bb.z;
  h.w = ag.w * id + rr.w + bb.w;
  if (do_bn_relu) {
    const float4 mu = reinterpret_cast<const float4*>(mean)[q];
    const float4 vv = reinterpret_cast<const float4*>(var)[q];
    const float4 gg = reinterpret_cast<const float4*>(gamma)[q];
    const float4 be = reinterpret_cast<const float4*>(beta)[q];
    h.x = fmaxf((h.x - mu.x) * rsqrtf(vv.x + 1e-5f) * gg.x + be.x, 0.f);
    h.y = fmaxf((h.y - mu.y) * rsqrtf(vv.y + 1e-5f) * gg.y + be.y, 0.f);
    h.z = fmaxf((h.z - mu.z) * rsqrtf(vv.z + 1e-5f) * gg.z + be.z, 0.f);
    h.w = fmaxf((h.w - mu.w) * rsqrtf(vv.w + 1e-5f) * gg.w + be.w, 0.f);
  }
  reinterpret_cast<float4*>(out)[idx] = h;
}

// Global mean pool, stage 1: atomic sums per graph (+ node counts).
__global__ void pool_accum(const int* __restrict__ batch,
                           const float* __restrict__ H,
                           float* __restrict__ outsum, float* __restrict__ cnt,
                           int n_nodes) {
  long long idx = (long long)blockIdx.x * blockDim.x + threadIdx.x;
  long long total = (long long)n_nodes * 32;  // 32 float4 chunks of 128
  if (idx >= total) return;
  const int node = (int)(idx >> 5);
  const int q = (int)(idx & 31);
  const int g = batch[node];
  const float4 v = *reinterpret_cast<const float4*>(H + (size_t)node * OUT_DIM +
                                                    (size_t)q * 4);
  float* o = outsum + (size_t)g * OUT_DIM + (size_t)q * 4;
  atomicAdd(o + 0, v.x);
  atomicAdd(o + 1, v.y);
  atomicAdd(o + 2, v.z);
  atomicAdd(o + 3, v.w);
  if (q == 0) atomicAdd(&cnt[g], 1.0f);
}

__global__ void pool_final(float* __restrict__ outsum,
                           const float* __restrict__ cnt) {
  int idx = blockIdx.x * blockDim.x + threadIdx.x;
  if (idx < G_GRAPHS * OUT_DIM)
    outsum[idx] /= fmaxf(cnt[idx >> 7], 1.0f);
}

// ---------------------------------------------------------------------------
extern "C" void kernel_launch(void* const* d_in, const int* in_sizes, int n_in,
                              void* d_out, int out_size, void* d_ws,
                              size_t ws_size, hipStream_t stream) {
  const float* x    = (const float*)d_in[0];
  const int*   eidx = (const int*)d_in[1];
  const int*   batch = (const int*)d_in[2];
  const float* W0l = (const float*)d_in[3];
  const float* W0r = (const float*)d_in[4];
  const float* b0  = (const float*)d_in[5];
  const float* W1l = (const float*)d_in[6];
  const float* W1r = (const float*)d_in[7];
  const float* b1  = (const float*)d_in[8];
  const float* W2l = (const float*)d_in[9];
  const float* W2r = (const float*)d_in[10];
  const float* b2  = (const float*)d_in[11];
  const float* bn0g = (const float*)d_in[12];
  const float* bn0b = (const float*)d_in[13];
  const float* bn0m = (const float*)d_in[14];
  const float* bn0v = (const float*)d_in[15];
  const float* bn1g = (const float*)d_in[16];
  const float* bn1b = (const float*)d_in[17];
  const float* bn1m = (const float*)d_in[18];
  const float* bn1v = (const float*)d_in[19];

  const int E = in_sizes[1] / 2;
  const int* esrc = eidx;
  const int* edst = eidx + E;
  const int nn = N_NODES;

  // Workspace layout (floats): ~102.6 MB total.
  float* ws = (float*)d_ws;
  float* invdeg = ws;                              // N
  float* Pb  = invdeg + nn;                        // N*128
  float* Rb  = Pb + (size_t)nn * 128;              // N*128
  float* Agg = Rb + (size_t)nn * 128;              // N*128
  float* H0  = Agg + (size_t)nn * 128;             // N*64
  float* H1  = H0 + (size_t)nn * 64;               // N*64
  float* cnt = H1 + (size_t)nn * 64;               // G

  float* out = (float*)d_out;

  const int gemmBlocks = nn / 16;  // 3125, exact
  const dim3 b256(256);

  // --- degree -> inverse degree ---
  zero_f32<<<512, 256, 0, stream>>>(invdeg, nn);
  deg_accum<<<(E + 255) / 256, b256, 0, stream>>>(edst, invdeg, E);
  deg_invert<<<(nn + 255) / 256, b256, 0, stream>>>(invdeg, nn);

  // --- layer 0: 128 -> 64 ---
  sage_dual_gemm<128, 64><<<gemmBlocks, 128, 0, stream>>>(x, W0l, W0r, Pb, Rb);
  zero_f32<<<2048, 256, 0, stream>>>(Agg, (long long)nn * 64);
  {
    long long tot = (long long)E * 16;  // F/4 = 16
    scatter_accum<<<(unsigned)((tot + 255) / 256), b256, 0, stream>>>(
        esrc, edst, Pb, Agg, E, 4);
  }
  ew_bn_relu4<<<(unsigned)(((long long)nn * 16 + 255) / 256), b256, 0,
                stream>>>(Agg, Rb, invdeg, b0, bn0g, bn0b, bn0m, bn0v, H0, nn,
                          4, 1);

  // --- layer 1: 64 -> 64 ---
  sage_dual_gemm<64, 64><<<gemmBlocks, 128, 0, stream>>>(H0, W1l, W1r, Pb, Rb);
  zero_f32<<<2048, 256, 0, stream>>>(Agg, (long long)nn * 64);
  {
    long long tot = (long long)E * 16;
    scatter_accum<<<(unsigned)((tot + 255) / 256), b256, 0, stream>>>(
        esrc, edst, Pb, Agg, E, 4);
  }
  ew_bn_relu4<<<(unsigned)(((long long)nn * 16 + 255) / 256), b256, 0,
                stream>>>(Agg, Rb, invdeg, b1, bn1g, bn1b, bn1m, bn1v, H1, nn,
                          4, 1);

  // --- layer 2: 64 -> 128 (no BN/ReLU) ---
  sage_dual_gemm<64, 128><<<gemmBlocks, 256, 0, stream>>>(H1, W2l, W2r, Pb, Rb);
  zero_f32<<<2048, 256, 0, stream>>>(Agg, (long long)nn * 128);
  {
    long long tot = (long long)E * 32;  // F/4 = 32
    scatter_accum<<<(unsigned)((tot + 255) / 256), b256, 0, stream>>>(
        esrc, edst, Pb, Agg, E, 5);
  }
  // H2 written into Pb (P no longer needed).
  ew_bn_relu4<<<(unsigned)(((long long)nn * 32 + 255) / 256), b256, 0,
                stream>>>(Agg, Rb, invdeg, b2, nullptr, nullptr, nullptr,
                          nullptr, Pb, nn, 5, 0);

  // --- global mean pool ---
  zero_f32<<<32, 256, 0, stream>>>(out, (long long)G_GRAPHS * OUT_DIM);
  zero_f32<<<1, 64, 0, stream>>>(cnt, G_GRAPHS);
  {
    long long tot = (long long)nn * 32;
    pool_accum<<<(unsigned)((tot + 255) / 256), b256, 0, stream>>>(batch, Pb,
                                                                   out, cnt,
                                                                   nn);
  }
  pool_final<<<(G_GRAPHS * OUT_DIM + 255) / 256, b256, 0, stream>>>(out, cnt);
}